// DCCA_2D_Loss_17480516895236
// MI455X (gfx1250) — compile-verified
//
#include <hip/hip_runtime.h>
#include <math.h>

// ---------------------------------------------------------------------------
// DCCA 2D loss on MI455X (gfx1250, wave32).
//
// Phase 1: three 64x64 fp32 Gram matrices over 262,144 samples (134 MB read
//          once -> ~5.8us HBM floor at 23.3 TB/s) on V_WMMA_F32_16X16X4_F32.
//          512 blocks x 4 waves; per-block partial Grams written to scratch
//          (plain stores), then a deterministic fixed-order serial reduce. No FP atomics.
// Phase 2: corr^2 = tr(S11^-1 S12 S22^-1 S12^T) via WMMA Newton-Schulz
//          inverses in a single workgroup (X0 = I/tr is provably convergent:
//          lambda_max <= tr for SPD, ridge bounds lambda_min > 0).
// ---------------------------------------------------------------------------

typedef float v2f __attribute__((ext_vector_type(2)));
typedef float v8f __attribute__((ext_vector_type(8)));

#define TSTR  68    // padded LDS tile stride (floats): 272B rows, 16B aligned
#define NPAIR 10    // upper-triangle 16x16 block pairs of a symmetric 64x64
#define NBLK  512   // gram grid size (fills ~256 WGPs at 2 waves/SIMD)
#define SLICES_PER_BLK 4

static constexpr int PI[NPAIR] = {0, 0, 0, 0, 1, 1, 1, 2, 2, 3};
static constexpr int PJ[NPAIR] = {0, 1, 2, 3, 1, 2, 3, 2, 3, 3};

__device__ __forceinline__ v8f wmma4(v2f a, v2f b, v8f c) {
  // D(16x16 f32) = A(16x4 f32) x B(4x16 f32) + C
  return __builtin_amdgcn_wmma_f32_16x16x4_f32(false, a, false, b, (short)0, c,
                                               false, false);
}

// A-fragment of M[rowBase:+16, kBase:+4] (row-major, stride ldm floats).
// Lane(g = lane>>4, t = lane&15) holds {M[rowBase+t][kBase+2g], [..+2g+1]}.
// Same layout is the B-fragment of (that block)^T, so for SYMMETRIC M this
// doubles as the B-fragment of M[kBase:+4, rowBase:+16].
__device__ __forceinline__ v2f frag_row(const float* M, int ldm, int rowBase,
                                        int kBase, int grp, int tl) {
  return *(const v2f*)(M + (rowBase + tl) * ldm + kBase + 2 * grp);
}

// B-fragment of M[kBase:+4, colBase:+16] for general (non-symmetric) M, ld=64.
__device__ __forceinline__ v2f frag_col(const float* M, int kBase, int colBase,
                                        int grp, int tl) {
  v2f r;
  r.x = M[(kBase + 2 * grp) * 64 + colBase + tl];
  r.y = M[(kBase + 2 * grp + 1) * 64 + colBase + tl];
  return r;
}

// ---------------------------------------------------------------------------
// Grid: NBLK blocks x 128 threads (4 waves); each block does 4 slices (64x128)
// and writes its partial {G11|G22|G12} (12288 floats) to P[block].
__global__ void __launch_bounds__(128) gram_kernel(const float* __restrict__ H1,
                                                   const float* __restrict__ H2,
                                                   float* __restrict__ P) {
  __shared__ __align__(16) float smem[12288];  // 48 KB: tiles, then reduction
  float* t1 = smem;                 // 64 x TSTR
  float* t2 = smem + 64 * TSTR;     // 64 x TSTR

  const int tid  = threadIdx.x;
  const int wave = tid >> 5;
  const int lane = tid & 31;
  const int grp  = lane >> 4;
  const int tl   = lane & 15;

  v8f acc11[NPAIR], acc22[NPAIR], acc12[4][4];
  const v8f vz = {0.f, 0.f, 0.f, 0.f, 0.f, 0.f, 0.f, 0.f};
#pragma unroll
  for (int p = 0; p < NPAIR; ++p) { acc11[p] = vz; acc22[p] = vz; }
#pragma unroll
  for (int bi = 0; bi < 4; ++bi)
#pragma unroll
    for (int bj = 0; bj < 4; ++bj) acc12[bi][bj] = vz;

  const int slice0 = blockIdx.x * SLICES_PER_BLK;
  for (int ss = 0; ss < SLICES_PER_BLK; ++ss) {
    const size_t base = (size_t)(slice0 + ss) * 64 * 128;
    for (int h = 0; h < 2; ++h) {            // k-halves of the slice
      __syncthreads();                       // previous tile fully consumed
      // Stage 64x64 of H1 and H2 into LDS: 2048 float4s, fully coalesced.
#pragma unroll
      for (int it = 0; it < 16; ++it) {
        const int fid = it * 128 + tid;      // 0..2047
        const int r   = (fid >> 4) & 63;
        const int c4  = fid & 15;
        const float* src = (fid < 1024) ? H1 : H2;
        float*       dst = (fid < 1024) ? t1 : t2;
        const float4 v = *(const float4*)(src + base + r * 128 + h * 64 + c4 * 4);
        *(float4*)(dst + r * TSTR + c4 * 4) = v;
      }
      __syncthreads();

      // Each wave contracts 4 K-positions (K=4 each) of this half.
#pragma unroll 1
      for (int q = 0; q < 4; ++q) {
        const int k0 = wave * 16 + q * 4;
        v2f f1[4], f2[4];
#pragma unroll
        for (int r = 0; r < 4; ++r) {
          f1[r] = frag_row(t1, TSTR, r * 16, k0, grp, tl);
          f2[r] = frag_row(t2, TSTR, r * 16, k0, grp, tl);
        }
#pragma unroll
        for (int p = 0; p < NPAIR; ++p) {    // symmetric grams: upper blocks
          acc11[p] = wmma4(f1[PI[p]], f1[PJ[p]], acc11[p]);
          acc22[p] = wmma4(f2[PI[p]], f2[PJ[p]], acc22[p]);
        }
#pragma unroll
        for (int bi = 0; bi < 4; ++bi)
#pragma unroll
          for (int bj = 0; bj < 4; ++bj)
            acc12[bi][bj] = wmma4(f1[bi], f2[bj], acc12[bi][bj]);
      }
    }
  }

  // ---- deterministic in-block combine: barrier-sequenced wave turns -------
  // Each wave's accumulators cover every one of the 12288 cells exactly once
  // (10 direct + 6 mirrored blocks per symmetric gram, 16 blocks for G12),
  // so wave 0 stores and waves 1..3 read-add-store in fixed order.
  __syncthreads();
  for (int wturn = 0; wturn < 4; ++wturn) {
    if (wave == wturn) {
      const bool first = (wturn == 0);
#pragma unroll
      for (int p = 0; p < NPAIR; ++p) {
        const int bi = PI[p], bj = PJ[p];
#pragma unroll
        for (int v = 0; v < 8; ++v) {
          const int r = bi * 16 + v + 8 * grp;  // C/D layout: M = vgpr + 8*grp
          const int c = bj * 16 + tl;           //             N = tl
          const int i11 = 0 * 4096 + r * 64 + c;
          const int i22 = 1 * 4096 + r * 64 + c;
          smem[i11] = first ? acc11[p][v] : (smem[i11] + acc11[p][v]);
          smem[i22] = first ? acc22[p][v] : (smem[i22] + acc22[p][v]);
          if (bi != bj) {                       // mirror off-diagonal blocks
            const int m11 = 0 * 4096 + c * 64 + r;
            const int m22 = 1 * 4096 + c * 64 + r;
            smem[m11] = first ? acc11[p][v] : (smem[m11] + acc11[p][v]);
            smem[m22] = first ? acc22[p][v] : (smem[m22] + acc22[p][v]);
          }
        }
      }
#pragma unroll
      for (int bi = 0; bi < 4; ++bi)
#pragma unroll
        for (int bj = 0; bj < 4; ++bj)
#pragma unroll
          for (int v = 0; v < 8; ++v) {
            const int r = bi * 16 + v + 8 * grp;
            const int c = bj * 16 + tl;
            const int i12 = 2 * 4096 + r * 64 + c;
            smem[i12] = first ? acc12[bi][bj][v] : (smem[i12] + acc12[bi][bj][v]);
          }
    }
    __syncthreads();
  }

  // ---- write this block's partial Gram: plain coalesced b128 stores -------
  float4* Pout = (float4*)(P + (size_t)blockIdx.x * 12288);
  const float4* s4 = (const float4*)smem;
  for (int j = tid; j < 3072; j += 128) Pout[j] = s4[j];
}

// ---------------------------------------------------------------------------
// Deterministic cross-block reduce: G[i] = sum_b P[b][i], fixed order.
__global__ void __launch_bounds__(256) reduce_kernel(const float* __restrict__ P,
                                                     float* __restrict__ G) {
  const int i = blockIdx.x * 256 + threadIdx.x;  // 0..12287 (coalesced reads)
  float s = 0.0f;
  for (int b = 0; b < NBLK; ++b) s += P[(size_t)b * 12288 + i];
  G[i] = s;
}

// ---------------------------------------------------------------------------
// One workgroup, 128 threads (4 waves, one 16-row stripe each).
__global__ void __launch_bounds__(128) solve_kernel(const float* __restrict__ G,
                                                    float* __restrict__ out) {
  __shared__ __align__(16) float S[4 * 4096];  // 64 KB: Sigma | X1 | X2 | T/S12
  float* RS  = S;            // current Sigma, later M2
  float* RX1 = S + 4096;     // Sigma11^-1
  float* RX2 = S + 8192;     // Sigma22^-1
  float* RT  = S + 12288;    // Newton temp T, later Sigma12

  const int tid  = threadIdx.x;
  const int w    = tid >> 5;
  const int lane = tid & 31;
  const int grp  = lane >> 4;
  const int tl   = lane & 15;

  const float m   = 2048.0f;                   // B*C
  const float scl = 1.0f - 1.0f / m;
  const float cf  = scl * scl / (m * (m - 1.0f));
  const v8f   vz  = {0.f, 0.f, 0.f, 0.f, 0.f, 0.f, 0.f, 0.f};

  // ---- Newton-Schulz SPD inverses: Sigma11 -> RX1, Sigma22 -> RX2 ----
  for (int mat = 0; mat < 2; ++mat) {
    float* RX = (mat == 0) ? RX1 : RX2;
    for (int i = tid; i < 4096; i += 128) {
      float v = cf * G[mat * 4096 + i];
      if ((i & 63) == (i >> 6)) v += 1e-4f;    // ridge on the diagonal
      RS[i] = v;
    }
    __syncthreads();
    if (tid == 0) {                            // deterministic serial trace
      float t = 0.0f;
      for (int r = 0; r < 64; ++r) t += RS[r * 65];
      RT[0] = t;
    }
    __syncthreads();
    const float inv_tr = 1.0f / RT[0];
    for (int i = tid; i < 4096; i += 128)
      RX[i] = ((i & 63) == (i >> 6)) ? inv_tr : 0.0f;  // X0 = I/tr (safe)
    __syncthreads();

    for (int it = 0; it < 14; ++it) {
      // T(stripe) = 2I - Sigma[stripe,:] @ X   (Sigma, X symmetric)
      v8f ta[4] = {vz, vz, vz, vz};
#pragma unroll 1
      for (int k = 0; k < 16; ++k) {
        v2f a = frag_row(RS, 64, w * 16, k * 4, grp, tl);
#pragma unroll
        for (int j = 0; j < 4; ++j) {
          v2f b = frag_row(RX, 64, j * 16, k * 4, grp, tl);  // B-frag via symmetry
          ta[j] = wmma4(a, b, ta[j]);
        }
      }
#pragma unroll
      for (int j = 0; j < 4; ++j)
#pragma unroll
        for (int v = 0; v < 8; ++v) {
          const int r = w * 16 + v + 8 * grp, c = j * 16 + tl;
          float x = -ta[j][v];
          if (r == c) x += 2.0f;
          RT[r * 64 + c] = x;                  // own stripe: no barrier needed
        }
      // Xnew(stripe) = T(stripe) @ X
      v8f xa[4] = {vz, vz, vz, vz};
#pragma unroll 1
      for (int k = 0; k < 16; ++k) {
        v2f a = frag_row(RT, 64, w * 16, k * 4, grp, tl);
#pragma unroll
        for (int j = 0; j < 4; ++j) {
          v2f b = frag_row(RX, 64, j * 16, k * 4, grp, tl);
          xa[j] = wmma4(a, b, xa[j]);
        }
      }
      __syncthreads();                         // all reads of old X done
#pragma unroll
      for (int j = 0; j < 4; ++j)
#pragma unroll
        for (int v = 0; v < 8; ++v)
          RX[(w * 16 + v + 8 * grp) * 64 + j * 16 + tl] = xa[j][v];
      __syncthreads();
    }
  }

  // ---- corr^2 = tr( (X1 @ S12) @ (X2 @ S12^T) ) ----
  for (int i = tid; i < 4096; i += 128) RT[i] = cf * G[2 * 4096 + i];  // S12
  __syncthreads();

  // M2 = X2 @ S12^T : B-frag of S12^T(k, j) == row-frag of S12 at (j, k)
  v8f m2[4] = {vz, vz, vz, vz};
#pragma unroll 1
  for (int k = 0; k < 16; ++k) {
    v2f a = frag_row(RX2, 64, w * 16, k * 4, grp, tl);
#pragma unroll
    for (int j = 0; j < 4; ++j) {
      v2f b = frag_row(RT, 64, j * 16, k * 4, grp, tl);
      m2[j] = wmma4(a, b, m2[j]);
    }
  }
  __syncthreads();                             // Sigma (RS) no longer needed
#pragma unroll
  for (int j = 0; j < 4; ++j)
#pragma unroll
    for (int v = 0; v < 8; ++v)
      RS[(w * 16 + v + 8 * grp) * 64 + j * 16 + tl] = m2[j][v];  // M2 -> RS
  __syncthreads();

  // M1(stripe) = X1 @ S12, then partial = sum M1[r,c] * M2[c,r]
  v8f m1[4] = {vz, vz, vz, vz};
#pragma unroll 1
  for (int k = 0; k < 16; ++k) {
    v2f a = frag_row(RX1, 64, w * 16, k * 4, grp, tl);
#pragma unroll
    for (int j = 0; j < 4; ++j) {
      v2f b = frag_col(RT, k * 4, j * 16, grp, tl);  // general B-frag of S12
      m1[j] = wmma4(a, b, m1[j]);
    }
  }
  float part = 0.0f;
#pragma unroll
  for (int j = 0; j < 4; ++j)
#pragma unroll
    for (int v = 0; v < 8; ++v) {
      const int r = w * 16 + v + 8 * grp, c = j * 16 + tl;
      part += m1[j][v] * RS[c * 64 + r];
    }
#pragma unroll
  for (int off = 16; off > 0; off >>= 1) part += __shfl_down(part, off, 32);
  __syncthreads();
  if (lane == 0) RX1[w] = part;                // X1 fully consumed above
  __syncthreads();
  if (tid == 0) out[0] = -sqrtf(RX1[0] + RX1[1] + RX1[2] + RX1[3]);
}

// ---------------------------------------------------------------------------
extern "C" void kernel_launch(void* const* d_in, const int* in_sizes, int n_in,
                              void* d_out, int out_size, void* d_ws,
                              size_t ws_size, hipStream_t stream) {
  (void)in_sizes; (void)n_in; (void)out_size; (void)ws_size;
  const float* H1 = (const float*)d_in[0];
  const float* H2 = (const float*)d_in[1];
  float* P   = (float*)d_ws;                    // NBLK x 12288 partial Grams
  float* G   = P + (size_t)NBLK * 12288;        // 12288: G11 | G22 | G12
  float* out = (float*)d_out;                   // 1 float

  gram_kernel<<<NBLK, 128, 0, stream>>>(H1, H2, P);
  reduce_kernel<<<48, 256, 0, stream>>>(P, G);
  solve_kernel<<<1, 128, 0, stream>>>(G, out);
}